// Scatter_1039382086096
// MI455X (gfx1250) — compile-verified
//
#include <hip/hip_runtime.h>
#include <cstdint>
#include <cstddef>

// Problem constants (match reference)
#define NNODES 20000
#define NCH    64
#define NEDGE  640000
#define NBATCH 8
#define FEAT   704        // 11 * 64
#define OUTROW 2816       // 4 * 704
#define OUTN   (NBATCH*OUTROW)

typedef __attribute__((ext_vector_type(2))) float v2f;
typedef __attribute__((ext_vector_type(8))) float v8f;

// gfx1250 async global->LDS staging (guarded so host pass / older toolchains still compile)
#if defined(__has_builtin)
#if __has_builtin(__builtin_amdgcn_global_load_async_to_lds_b64) && \
    __has_builtin(__builtin_amdgcn_s_wait_asynccnt)
#define USE_ASYNC 1
#endif
#endif
#ifndef USE_ASYNC
#define USE_ASYNC 0
#endif

#if USE_ASYNC
// builtin signature (from hipcc diagnostic): (v2i AS1*, v2i AS3*, imm int, imm int)
typedef int async_v2i __attribute__((vector_size(8)));
#define ASYNC_COPY_B64(SRC, DST)                                        \
  __builtin_amdgcn_global_load_async_to_lds_b64(                        \
      (__attribute__((address_space(1))) async_v2i*)(SRC),              \
      (__attribute__((address_space(3))) async_v2i*)(DST), 0, 0)
#endif

// ---------------------------------------------------------------- setup ----

__global__ void degree_kernel(const int* __restrict__ col, int* __restrict__ counts, int e) {
  int i = blockIdx.x * blockDim.x + threadIdx.x;
  if (i < e) atomicAdd(&counts[col[i]], 1);
}

__global__ void dinv_kernel(const int* __restrict__ counts, float* __restrict__ dinv, int n) {
  int i = blockIdx.x * blockDim.x + threadIdx.x;
  if (i < n) dinv[i] = rsqrtf((float)counts[i] + 1.0f);  // +1 self loop; deg >= 1
}

// single-block chunked inclusive scan -> exclusive offsets[0..n]
__global__ void scan_kernel(const int* __restrict__ counts, int* __restrict__ offsets, int n) {
  __shared__ int buf[1024];
  __shared__ int carry;
  if (threadIdx.x == 0) carry = 0;
  __syncthreads();
  for (int base = 0; base < n; base += 1024) {
    int i = base + (int)threadIdx.x;
    int v = (i < n) ? counts[i] : 0;
    buf[threadIdx.x] = v;
    __syncthreads();
    for (int off = 1; off < 1024; off <<= 1) {
      int t = ((int)threadIdx.x >= off) ? buf[threadIdx.x - off] : 0;
      __syncthreads();
      buf[threadIdx.x] += t;
      __syncthreads();
    }
    int incl = buf[threadIdx.x] + carry;
    if (i < n) offsets[i + 1] = incl;
    __syncthreads();
    if (threadIdx.x == 1023) carry += buf[1023];
    __syncthreads();
  }
  if (threadIdx.x == 0) offsets[0] = 0;
}

// packed CSR entry: .x = source row, .y = bitcast(norm)
__global__ void fill_csr_kernel(const int* __restrict__ row, const int* __restrict__ col,
                                const int* __restrict__ offs, int* __restrict__ cursor,
                                const float* __restrict__ dinv,
                                int2* __restrict__ epack, int e) {
  int i = blockIdx.x * blockDim.x + threadIdx.x;
  if (i < e) {
    int c = col[i], r = row[i];
    int p = offs[c] + atomicAdd(&cursor[c], 1);
    int2 ev;
    ev.x = r;
    ev.y = __float_as_int(dinv[r] * dinv[c]);
    epack[p] = ev;
  }
}

__global__ void cntb_kernel(const int* __restrict__ batch, int* __restrict__ cntb, int n) {
  int i = blockIdx.x * blockDim.x + threadIdx.x;
  if (i < n) atomicAdd(&cntb[batch[i]], 1);
}

// feats row 0 = x, and h0 of level-1 cascade = x
__global__ void copyx_kernel(const float* __restrict__ x, float* __restrict__ feats,
                             float* __restrict__ h, int total) {
  int i = blockIdx.x * blockDim.x + threadIdx.x;
  if (i < total) {
    int n = i / NCH, c = i % NCH;
    float v = x[i];
    feats[(size_t)n * FEAT + c] = v;
    h[i] = v;
  }
}

// ------------------------------------------------------------- cascades ----
// One wave per node. Edge metadata is staged 32-at-a-time into LDS with
// GLOBAL_LOAD_ASYNC_TO_LDS_B64 (double-buffered, s_wait_asynccnt), then
// consumed via LDS broadcast reads while the next chunk is in flight.
// h_new[v] = 0.5*(h[v] + dinv[v]^2*h[v] + sum_in norm*h[src])

__global__ __launch_bounds__(32)
void cascade64_kernel(const float* __restrict__ h, float* __restrict__ hn,
                      const int* __restrict__ offs, const int2* __restrict__ epack,
                      const float* __restrict__ dinv) {
  int v = blockIdx.x, lane = threadIdx.x;
  float dv = dinv[v];
  float2 hv = ((const float2*)(h + (size_t)v * NCH))[lane];
  float2 acc;
  acc.x = dv * dv * hv.x;
  acc.y = dv * dv * hv.y;
  int beg = offs[v], end = offs[v + 1];
  int cnt = end - beg;
#if USE_ASYNC
  __shared__ int2 ebuf[2][32];
  {
    int rem = cnt < 32 ? cnt : 32;
    if (lane < rem)
      ASYNC_COPY_B64(epack + beg + lane, &ebuf[0][lane]);
  }
  int nch = (cnt + 31) >> 5;
  for (int ch = 0; ch < nch; ++ch) {
    int cbase = ch << 5;
    int clen = cnt - cbase; if (clen > 32) clen = 32;
    __builtin_amdgcn_s_wait_asynccnt(0);
    if (cbase + 32 < cnt) {
      int rem = cnt - cbase - 32; if (rem > 32) rem = 32;
      if (lane < rem)
        ASYNC_COPY_B64(epack + beg + cbase + 32 + lane, &ebuf[(ch + 1) & 1][lane]);
    }
    const int2* eb = ebuf[ch & 1];
    for (int j = 0; j < clen; ++j) {
      int2 e = eb[j];                                   // LDS broadcast
      float w = __int_as_float(e.y);
      float2 hr = ((const float2*)(h + (size_t)e.x * NCH))[lane];
      acc.x = fmaf(w, hr.x, acc.x);
      acc.y = fmaf(w, hr.y, acc.y);
    }
  }
#else
  for (int j = beg; j < end; ++j) {
    if (((j - beg) & 31) == 0 && j + 64 < end)
      __builtin_prefetch(epack + j + 64, 0, 1);
    int2 e = epack[j];
    float w = __int_as_float(e.y);
    float2 hr = ((const float2*)(h + (size_t)e.x * NCH))[lane];
    acc.x = fmaf(w, hr.x, acc.x);
    acc.y = fmaf(w, hr.y, acc.y);
  }
#endif
  float2 o;
  o.x = 0.5f * (hv.x + acc.x);
  o.y = 0.5f * (hv.y + acc.y);
  ((float2*)(hn + (size_t)v * NCH))[lane] = o;
}

__global__ __launch_bounds__(32)
void cascade256_kernel(const float* __restrict__ h, float* __restrict__ hn,
                       const int* __restrict__ offs, const int2* __restrict__ epack,
                       const float* __restrict__ dinv) {
  int v = blockIdx.x, lane = threadIdx.x;
  float dv = dinv[v];
  const float4* hrow = (const float4*)(h + (size_t)v * 256);
  float4 hva = hrow[lane * 2];
  float4 hvb = hrow[lane * 2 + 1];
  float sl = dv * dv;
  float4 aa, ab;
  aa.x = sl * hva.x; aa.y = sl * hva.y; aa.z = sl * hva.z; aa.w = sl * hva.w;
  ab.x = sl * hvb.x; ab.y = sl * hvb.y; ab.z = sl * hvb.z; ab.w = sl * hvb.w;
  int beg = offs[v], end = offs[v + 1];
  int cnt = end - beg;
#if USE_ASYNC
  __shared__ int2 ebuf[2][32];
  {
    int rem = cnt < 32 ? cnt : 32;
    if (lane < rem)
      ASYNC_COPY_B64(epack + beg + lane, &ebuf[0][lane]);
  }
  int nch = (cnt + 31) >> 5;
  for (int ch = 0; ch < nch; ++ch) {
    int cbase = ch << 5;
    int clen = cnt - cbase; if (clen > 32) clen = 32;
    __builtin_amdgcn_s_wait_asynccnt(0);
    if (cbase + 32 < cnt) {
      int rem = cnt - cbase - 32; if (rem > 32) rem = 32;
      if (lane < rem)
        ASYNC_COPY_B64(epack + beg + cbase + 32 + lane, &ebuf[(ch + 1) & 1][lane]);
    }
    const int2* eb = ebuf[ch & 1];
    for (int j = 0; j < clen; ++j) {
      int2 e = eb[j];                                   // LDS broadcast
      float w = __int_as_float(e.y);
      const float4* rr = (const float4*)(h + (size_t)e.x * 256);
      float4 ra = rr[lane * 2];
      float4 rb = rr[lane * 2 + 1];
      aa.x = fmaf(w, ra.x, aa.x); aa.y = fmaf(w, ra.y, aa.y);
      aa.z = fmaf(w, ra.z, aa.z); aa.w = fmaf(w, ra.w, aa.w);
      ab.x = fmaf(w, rb.x, ab.x); ab.y = fmaf(w, rb.y, ab.y);
      ab.z = fmaf(w, rb.z, ab.z); ab.w = fmaf(w, rb.w, ab.w);
    }
  }
#else
  for (int j = beg; j < end; ++j) {
    if (((j - beg) & 31) == 0 && j + 64 < end)
      __builtin_prefetch(epack + j + 64, 0, 1);
    int2 e = epack[j];
    float w = __int_as_float(e.y);
    const float4* rr = (const float4*)(h + (size_t)e.x * 256);
    float4 ra = rr[lane * 2];
    float4 rb = rr[lane * 2 + 1];
    aa.x = fmaf(w, ra.x, aa.x); aa.y = fmaf(w, ra.y, aa.y);
    aa.z = fmaf(w, ra.z, aa.z); aa.w = fmaf(w, ra.w, aa.w);
    ab.x = fmaf(w, rb.x, ab.x); ab.y = fmaf(w, rb.y, ab.y);
    ab.z = fmaf(w, rb.z, ab.z); ab.w = fmaf(w, rb.w, ab.w);
  }
#endif
  float4 oa, ob;
  oa.x = 0.5f * (hva.x + aa.x); oa.y = 0.5f * (hva.y + aa.y);
  oa.z = 0.5f * (hva.z + aa.z); oa.w = 0.5f * (hva.w + aa.w);
  ob.x = 0.5f * (hvb.x + ab.x); ob.y = 0.5f * (hvb.y + ab.y);
  ob.z = 0.5f * (hvb.z + ab.z); ob.w = 0.5f * (hvb.w + ab.w);
  float4* orow = (float4*)(hn + (size_t)v * 256);
  orow[lane * 2]     = oa;
  orow[lane * 2 + 1] = ob;
}

// level-1 snapshot/emit: at step 2^(w+1) (w>=0): a=|h-snap|; feats row (1+w); seed level-2 h0 slot w.
// w<0: snapshot only (step 1).
__global__ void emit_l1(const float* __restrict__ h, float* __restrict__ snap,
                        float* __restrict__ feats, float* __restrict__ h2, int w, int total) {
  int i = blockIdx.x * blockDim.x + threadIdx.x;
  if (i < total) {
    int n = i / NCH, c = i % NCH;
    float hv = h[i];
    if (w >= 0) {
      float a = fabsf(hv - snap[i]);
      feats[(size_t)n * FEAT + (size_t)(1 + w) * NCH + c] = a;
      h2[(size_t)n * 256 + (size_t)c * 4 + w] = a;
    }
    snap[i] = hv;
  }
}

// level-2 snapshot/emit: features laid out f = c*4 + b. rowBase<0: snapshot only.
// emits feats rows rowBase..rowBase+nb-1 = |h - snap| for b = 0..nb-1.
__global__ void emit_l2(const float* __restrict__ h, float* __restrict__ snap,
                        float* __restrict__ feats, int rowBase, int nb, int total) {
  int i = blockIdx.x * blockDim.x + threadIdx.x;
  if (i < total) {
    int n = i / 256, r = i % 256;
    int c = r >> 2, b = r & 3;
    float hv = h[i];
    if (rowBase >= 0 && b < nb) {
      feats[(size_t)n * FEAT + (size_t)(rowBase + b) * NCH + c] = fabsf(hv - snap[i]);
    }
    snap[i] = hv;
  }
}

// ---------------------------------------------- WMMA segmented moments -----
// S_p[f, b] = sum_n d[n, f]^p * (batch[n] == b)  for p = 1..4
// One wave per 16-feature tile; K swept over nodes in chunks of 4 using
// V_WMMA_F32_16X16X4_F32 (full f32, no precision loss).
__global__ __launch_bounds__(32)
void moments_wmma(const float* __restrict__ d, const int* __restrict__ batch,
                  float* __restrict__ S1o, float* __restrict__ S2o,
                  float* __restrict__ S3o, float* __restrict__ S4o, int n) {
  int f0   = blockIdx.x * 16;
  int lane = threadIdx.x;
  int half = lane >> 4;     // 0: K pair {0,1}, 1: K pair {2,3}  (A & B layouts)
  int mn   = lane & 15;     // A row M / B col N for this lane
  int kb   = half * 2;

  v8f c1 = {}, c2 = {}, c3 = {}, c4 = {};

  for (int k0 = 0; k0 < n; k0 += 4) {           // n divisible by 4
    int ka = k0 + kb, kb1 = ka + 1;
    float x0 = d[(size_t)ka  * FEAT + f0 + mn];
    float x1 = d[(size_t)kb1 * FEAT + f0 + mn];
    int  b0 = batch[ka], b1 = batch[kb1];

    v2f a1 = {x0, x1};
    v2f a2 = {x0 * x0, x1 * x1};
    v2f a3 = {a2.x * x0, a2.y * x1};
    v2f a4 = {a2.x * a2.x, a2.y * a2.y};
    v2f bm = {(b0 == mn) ? 1.0f : 0.0f, (b1 == mn) ? 1.0f : 0.0f};

    c1 = __builtin_amdgcn_wmma_f32_16x16x4_f32(false, a1, false, bm, (short)0, c1, false, false);
    c2 = __builtin_amdgcn_wmma_f32_16x16x4_f32(false, a2, false, bm, (short)0, c2, false, false);
    c3 = __builtin_amdgcn_wmma_f32_16x16x4_f32(false, a3, false, bm, (short)0, c3, false, false);
    c4 = __builtin_amdgcn_wmma_f32_16x16x4_f32(false, a4, false, bm, (short)0, c4, false, false);
  }

  // C/D layout: VGPR r, lanes 0-15 -> (M=r, N=lane); lanes 16-31 -> (M=r+8, N=lane-16)
  if (mn < NBATCH) {
    for (int r = 0; r < 8; ++r) {
      int M = r + half * 8;
      int idx = (f0 + M) * NBATCH + mn;
      S1o[idx] = c1[r];
      S2o[idx] = c2[r];
      S3o[idx] = c3[r];
      S4o[idx] = c4[r];
    }
  }
}

// ------------------------------------------------------------- finalize ----
__global__ void finalize_kernel(const float* __restrict__ S1, const float* __restrict__ S2,
                                const float* __restrict__ S3, const float* __restrict__ S4,
                                const int* __restrict__ cntb, float* __restrict__ out) {
  int i = blockIdx.x * blockDim.x + threadIdx.x;
  if (i < OUTN) {
    int b = i / FEAT, f = i % FEAT;
    float cnt = fmaxf((float)cntb[b], 1.0f);
    int idx = f * NBATCH + b;
    float s1 = S1[idx], s2 = S2[idx], s3 = S3[idx], s4 = S4[idx];
    float mean = s1 / cnt;
    float e2 = s2 / cnt, e3 = s3 / cnt, e4 = s4 / cnt;
    float m2 = e2 - mean * mean;
    float m3 = e3 - 3.0f * mean * e2 + 2.0f * mean * mean * mean;
    float m4 = e4 - 4.0f * mean * e3 + 6.0f * mean * mean * e2
             - 3.0f * mean * mean * mean * mean;
    bool safe = m2 > 0.0f;
    float vs = safe ? m2 : 1.0f;
    float skew = safe ? (m3 / (vs * sqrtf(vs))) : 0.0f;
    float kurt = safe ? (m4 / (vs * vs)) : -3.0f;
    size_t ob = (size_t)b * OUTROW;
    out[ob + f]        = mean;
    out[ob + 704  + f] = m2;
    out[ob + 1408 + f] = skew;
    out[ob + 2112 + f] = kurt;
  }
}

__global__ void wavelet_copy_kernel(const float* __restrict__ w, float* __restrict__ out) {
  int i = threadIdx.x;
  if (i < 68) out[(size_t)OUTN + i] = w[i];
}

// ----------------------------------------------------------------- host ----

extern "C" void kernel_launch(void* const* d_in, const int* in_sizes, int n_in,
                              void* d_out, int out_size, void* d_ws, size_t ws_size,
                              hipStream_t stream) {
  const float* x       = (const float*)d_in[0];
  const int*   eidx    = (const int*)d_in[1];     // (2, E) row-major: row then col
  const int*   batch   = (const int*)d_in[2];
  const float* wavelet = (const float*)d_in[3];
  const int* erow = eidx;
  const int* ecol = eidx + NEDGE;
  float* out = (float*)d_out;

  // ----- carve workspace (aligned 256B) -----
  char* p = (char*)d_ws;
  auto carve = [&](size_t bytes) -> void* {
    void* r = (void*)p;
    p += (bytes + 255) & ~(size_t)255;
    return r;
  };
  int*   counts  = (int*)  carve((size_t)NNODES * 4);
  int*   offsets = (int*)  carve((size_t)(NNODES + 1) * 4);
  int*   cursor  = (int*)  carve((size_t)NNODES * 4);
  float* dinv    = (float*)carve((size_t)NNODES * 4);
  int2*  epack   = (int2*) carve((size_t)NEDGE * 8);
  float* h64a    = (float*)carve((size_t)NNODES * NCH * 4);
  float* h64b    = (float*)carve((size_t)NNODES * NCH * 4);
  float* snap1   = (float*)carve((size_t)NNODES * NCH * 4);
  float* h256a   = (float*)carve((size_t)NNODES * 256 * 4);
  float* h256b   = (float*)carve((size_t)NNODES * 256 * 4);
  float* snap2   = (float*)carve((size_t)NNODES * 256 * 4);
  float* feats   = (float*)carve((size_t)NNODES * FEAT * 4);
  float* S1      = (float*)carve((size_t)FEAT * NBATCH * 4);
  float* S2      = (float*)carve((size_t)FEAT * NBATCH * 4);
  float* S3      = (float*)carve((size_t)FEAT * NBATCH * 4);
  float* S4      = (float*)carve((size_t)FEAT * NBATCH * 4);
  int*   cntb    = (int*)  carve((size_t)NBATCH * 4);
  (void)ws_size; (void)in_sizes; (void)n_in; (void)out_size;

  // ----- graph setup: degrees, CSR (pull mode, by destination), batch counts -----
  (void)hipMemsetAsync(counts, 0, (size_t)NNODES * 4, stream);
  (void)hipMemsetAsync(cursor, 0, (size_t)NNODES * 4, stream);
  (void)hipMemsetAsync(cntb,   0, (size_t)NBATCH * 4, stream);

  degree_kernel<<<(NEDGE + 255) / 256, 256, 0, stream>>>(ecol, counts, NEDGE);
  dinv_kernel<<<(NNODES + 255) / 256, 256, 0, stream>>>(counts, dinv, NNODES);
  scan_kernel<<<1, 1024, 0, stream>>>(counts, offsets, NNODES);
  fill_csr_kernel<<<(NEDGE + 255) / 256, 256, 0, stream>>>(erow, ecol, offsets, cursor,
                                                           dinv, epack, NEDGE);
  cntb_kernel<<<(NNODES + 255) / 256, 256, 0, stream>>>(batch, cntb, NNODES);

  // ----- feats row 0 + level-1 h0 -----
  copyx_kernel<<<(NNODES * NCH + 255) / 256, 256, 0, stream>>>(x, feats, h64a, NNODES * NCH);

  // ----- level-1 cascade (width 64), snapshots at steps 1,2,4,8,16 -----
  {
    const float* hc = h64a; float* hn = h64b;
    for (int s = 1; s <= 16; ++s) {
      cascade64_kernel<<<NNODES, 32, 0, stream>>>(hc, hn, offsets, epack, dinv);
      const float* t = hn; hn = (float*)hc; hc = t;
      int w = (s == 1) ? -1 : (s == 2) ? 0 : (s == 4) ? 1 : (s == 8) ? 2 : (s == 16) ? 3 : -2;
      if (w != -2)
        emit_l1<<<(NNODES * NCH + 255) / 256, 256, 0, stream>>>((const float*)hc, snap1,
                                                                feats, h256a, w, NNODES * NCH);
    }
  }

  // ----- level-2 cascade (width 256 = c*4 + b), snapshots at 2,4,8,16 -----
  {
    const float* hc = h256a; float* hn = h256b;
    for (int s = 1; s <= 16; ++s) {
      cascade256_kernel<<<NNODES, 32, 0, stream>>>(hc, hn, offsets, epack, dinv);
      const float* t = hn; hn = (float*)hc; hc = t;
      if (s == 2)
        emit_l2<<<(NNODES * 256 + 255) / 256, 256, 0, stream>>>((const float*)hc, snap2,
                                                                feats, -1, 0, NNODES * 256);
      else if (s == 4)   // (w=1,b=0) -> row 5
        emit_l2<<<(NNODES * 256 + 255) / 256, 256, 0, stream>>>((const float*)hc, snap2,
                                                                feats, 5, 1, NNODES * 256);
      else if (s == 8)   // (w=2,b=0..1) -> rows 6,7
        emit_l2<<<(NNODES * 256 + 255) / 256, 256, 0, stream>>>((const float*)hc, snap2,
                                                                feats, 6, 2, NNODES * 256);
      else if (s == 16)  // (w=3,b=0..2) -> rows 8,9,10
        emit_l2<<<(NNODES * 256 + 255) / 256, 256, 0, stream>>>((const float*)hc, snap2,
                                                                feats, 8, 3, NNODES * 256);
    }
  }

  // ----- segmented raw moments via f32 WMMA (44 feature tiles) -----
  moments_wmma<<<FEAT / 16, 32, 0, stream>>>(feats, batch, S1, S2, S3, S4, NNODES);

  // ----- mean/var/skew/kurt + wavelet passthrough -----
  finalize_kernel<<<(OUTN + 255) / 256, 256, 0, stream>>>(S1, S2, S3, S4, cntb, out);
  wavelet_copy_kernel<<<1, 128, 0, stream>>>(wavelet, out);
}